// GraphCare_39067022524639
// MI455X (gfx1250) — compile-verified
//
#include <hip/hip_runtime.h>
#include <hip/hip_bf16.h>

typedef __attribute__((ext_vector_type(2))) float v2f;
typedef __attribute__((ext_vector_type(8))) float v8f;
typedef int v4i_vec __attribute__((vector_size(4 * sizeof(int))));
typedef __attribute__((address_space(1))) v4i_vec* gptr_b128;
typedef __attribute__((address_space(3))) v4i_vec* lptr_b128;

#define NUM_NODES 5000
#define MAX_VISIT 20
#define HID 128
#define OUT_DIM 25
#define LAYERS 3
#define BATCH 32
#define N_TOTAL 100000
#define E_TOTAL 400000
#define DECAY_RATE 0.01f
#define AKCHUNK 64
#define BPAD 4   // LDS row stride pad: 68 floats = 272B, 16B aligned

#if __has_builtin(__builtin_amdgcn_global_load_async_to_lds_b128)
#define HAVE_ASYNC_LDS 1
#else
#define HAVE_ASYNC_LDS 0
#endif

__device__ __forceinline__ void wait_async0() {
#if __has_builtin(__builtin_amdgcn_s_wait_asynccnt)
    __builtin_amdgcn_s_wait_asynccnt(0);
#else
    asm volatile("s_wait_asynccnt 0x0" ::: "memory");
#endif
}

// ---------------------------------------------------------------------------
// out[M,128] = gather(table, idx) @ W[128,128]^T + bias
// One 16x16 tile per wave via V_WMMA_F32_16X16X4_F32; no in-loop branches.
// A layout: lanes 0-15 hold M=lane, K={0,1}; lanes 16-31 hold M=lane-16, K={2,3}.
// ---------------------------------------------------------------------------
__global__ void __launch_bounds__(128)
gemm128_gather_wmma(const float* __restrict__ table, const int* __restrict__ idx,
                    const float* __restrict__ W, const float* __restrict__ bias,
                    float* __restrict__ out)
{
    const int lane = threadIdx.x & 31;
    const int wave = threadIdx.x >> 5;
    const int lo = lane & 15;
    const int hi = lane >> 4;
    const int row0 = blockIdx.x * 16;
    const int n0 = (blockIdx.y * 4 + wave) * 16;

    const float* arow = table + (size_t)idx[row0 + lo] * HID;
    const float* wrow = W + (size_t)(n0 + lo) * HID;

    v8f acc = {0.f, 0.f, 0.f, 0.f, 0.f, 0.f, 0.f, 0.f};
#pragma unroll
    for (int k0 = 0; k0 < HID; k0 += 4) {
        const int k = k0 + 2 * hi;
        float2 av = *(const float2*)(arow + k);
        float2 bv = *(const float2*)(wrow + k);
        v2f a; a.x = av.x; a.y = av.y;
        v2f b; b.x = bv.x; b.y = bv.y;
        acc = __builtin_amdgcn_wmma_f32_16x16x4_f32(false, a, false, b,
                                                    (short)0, acc, false, false);
    }
    const float bval = bias[n0 + lo];
#pragma unroll
    for (int r = 0; r < 8; ++r) {
        const int m = r + 8 * hi;
        out[(size_t)(row0 + m) * HID + n0 + lo] = acc[r] + bval;
    }
}

// ---------------------------------------------------------------------------
// out[M,128] = (A + A2) @ W[128,128]^T + bias      (conv layer: agg + x)
// ---------------------------------------------------------------------------
__global__ void __launch_bounds__(128)
gemm128_add_wmma(const float* __restrict__ A, const float* __restrict__ A2,
                 const float* __restrict__ W, const float* __restrict__ bias,
                 float* __restrict__ out)
{
    const int lane = threadIdx.x & 31;
    const int wave = threadIdx.x >> 5;
    const int lo = lane & 15;
    const int hi = lane >> 4;
    const int row0 = blockIdx.x * 16;
    const int n0 = (blockIdx.y * 4 + wave) * 16;

    const float* arow = A + (size_t)(row0 + lo) * HID;
    const float* a2row = A2 + (size_t)(row0 + lo) * HID;
    const float* wrow = W + (size_t)(n0 + lo) * HID;

    v8f acc = {0.f, 0.f, 0.f, 0.f, 0.f, 0.f, 0.f, 0.f};
#pragma unroll
    for (int k0 = 0; k0 < HID; k0 += 4) {
        const int k = k0 + 2 * hi;
        float2 av = *(const float2*)(arow + k);
        float2 t2 = *(const float2*)(a2row + k);
        float2 bv = *(const float2*)(wrow + k);
        v2f a; a.x = av.x + t2.x; a.y = av.y + t2.y;
        v2f b; b.x = bv.x; b.y = bv.y;
        acc = __builtin_amdgcn_wmma_f32_16x16x4_f32(false, a, false, b,
                                                    (short)0, acc, false, false);
    }
    const float bval = bias[n0 + lo];
#pragma unroll
    for (int r = 0; r < 8; ++r) {
        const int m = r + 8 * hi;
        out[(size_t)(row0 + m) * HID + n0 + lo] = acc[r] + bval;
    }
}

// ---------------------------------------------------------------------------
// alpha logits: out[640,5000] = visit[640,5000] @ W[5000,5000]^T + bias.
// 4 waves/WG covering 64 rows x 16 cols. B tile (16 n-rows x 64 k) staged in
// LDS via GLOBAL_LOAD_ASYNC_TO_LDS_B128 (ASYNCcnt) when available.
// K = 5000 = 78*64 + 8: full chunks are branch-free (16 WMMA each); the 8-wide
// tail uses a wave-uniform trip count so no EXEC masking touches the WMMAs.
// ---------------------------------------------------------------------------
__global__ void __launch_bounds__(128)
alpha_gemm_wmma(const float* __restrict__ visit,  // [B*V, 5000]
                const float* __restrict__ W,      // [5000, 5000] layer slice
                const float* __restrict__ bias,   // [5000] layer slice
                float* __restrict__ out)          // [B*V, 5000]
{
    __shared__ float sB[16][AKCHUNK + BPAD];  // row-major per n, stride 68
    const int t = threadIdx.x;
    const int lane = t & 31;
    const int wave = t >> 5;
    const int lo = lane & 15;
    const int hi = lane >> 4;
    const int row0 = blockIdx.x * 64 + wave * 16;
    const int n0 = blockIdx.y * 16;
    const int ncol = n0 + lo;

    const float* arow = visit + (size_t)(row0 + lo) * NUM_NODES;
    const int ln = t >> 3;   // 0..15: which B row (n) this thread stages
    const int kseg = t & 7;  // 0..7 : 8-float K segment within the chunk
    const bool nvalid = (n0 + ln) < NUM_NODES;
    const float* wrow = W + (size_t)(nvalid ? (n0 + ln) : 0) * NUM_NODES;
    float* ldst = &sB[ln][kseg * 8];

    v8f acc = {0.f, 0.f, 0.f, 0.f, 0.f, 0.f, 0.f, 0.f};

    const int KFULL = (NUM_NODES / AKCHUNK) * AKCHUNK;  // 4992
    for (int kk = 0; kk < KFULL; kk += AKCHUNK) {
        const float* gsrc = wrow + kk + kseg * 8;
        if (nvalid) {
#if HAVE_ASYNC_LDS
            __builtin_amdgcn_global_load_async_to_lds_b128(
                (gptr_b128)gsrc, (lptr_b128)ldst, 0, 0);
            __builtin_amdgcn_global_load_async_to_lds_b128(
                (gptr_b128)(gsrc + 4), (lptr_b128)(ldst + 4), 0, 0);
#else
#pragma unroll
            for (int j = 0; j < 8; ++j) ldst[j] = gsrc[j];
#endif
        } else {
#pragma unroll
            for (int j = 0; j < 8; ++j) ldst[j] = 0.f;
        }
#if HAVE_ASYNC_LDS
        wait_async0();
#endif
        if (kk + AKCHUNK < KFULL)
            __builtin_prefetch(wrow + kk + AKCHUNK, 0, 1);  // global_prefetch_b8
        __syncthreads();
#pragma unroll
        for (int k0 = 0; k0 < AKCHUNK; k0 += 4) {
            const int k = kk + k0 + 2 * hi;
            float2 av = *(const float2*)(arow + k);
            v2f a; a.x = av.x; a.y = av.y;
            v2f b;
            b.x = sB[lo][k0 + 2 * hi];
            b.y = sB[lo][k0 + 2 * hi + 1];
            acc = __builtin_amdgcn_wmma_f32_16x16x4_f32(false, a, false, b,
                                                        (short)0, acc, false, false);
        }
        __syncthreads();
    }

    // tail chunk: K = 5000-4992 = 8 (wave-uniform, 4-divisible)
    {
        const int kk = KFULL;
        const int ktail = NUM_NODES - KFULL;  // 8
#pragma unroll
        for (int j = 0; j < 8; ++j) {
            const int k = kk + kseg * 8 + j;
            ldst[j] = (nvalid && k < NUM_NODES) ? wrow[k] : 0.f;
        }
        __syncthreads();
        for (int k0 = 0; k0 < ktail; k0 += 4) {
            const int k = kk + k0 + 2 * hi;
            float2 av = *(const float2*)(arow + k);
            v2f a; a.x = av.x; a.y = av.y;
            v2f b;
            b.x = sB[lo][k0 + 2 * hi];
            b.y = sB[lo][k0 + 2 * hi + 1];
            acc = __builtin_amdgcn_wmma_f32_16x16x4_f32(false, a, false, b,
                                                        (short)0, acc, false, false);
        }
    }

    if (ncol < NUM_NODES) {
        const float bval = bias[ncol];
#pragma unroll
        for (int r = 0; r < 8; ++r) {
            const int m = r + 8 * hi;
            out[(size_t)(row0 + m) * NUM_NODES + ncol] = acc[r] + bval;
        }
    }
}

// ---------------------------------------------------------------------------
// beta[bv] = tanh(dot(beta_w, visit[bv]) + beta_b) * exp(gamma*(V - v))
// ---------------------------------------------------------------------------
__global__ void __launch_bounds__(32)
beta_kernel(const float* __restrict__ visit, const float* __restrict__ bw,
            const float* __restrict__ bb, float* __restrict__ beta_out)
{
    const int bv = blockIdx.x;
    const int lane = threadIdx.x;
    const float* vrow = visit + (size_t)bv * NUM_NODES;
    float s = 0.f;
    for (int k = lane; k < NUM_NODES; k += 32) s += vrow[k] * bw[k];
#pragma unroll
    for (int m = 16; m > 0; m >>= 1) s += __shfl_xor(s, m, 32);
    if (lane == 0) {
        const int v = bv % MAX_VISIT;
        beta_out[bv] = tanhf(s + bb[0]) * __expf(DECAY_RATE * (float)(MAX_VISIT - v));
    }
}

// ---------------------------------------------------------------------------
// attn[b,m] = sum_v softmax_v(logits[b,v,m]) * beta[b,v]
// ---------------------------------------------------------------------------
__global__ void __launch_bounds__(256)
attn_kernel(const float* __restrict__ logits, const float* __restrict__ beta,
            float* __restrict__ attn)
{
    const int t = blockIdx.x * blockDim.x + threadIdx.x;
    if (t >= BATCH * NUM_NODES) return;
    const int b = t / NUM_NODES;
    const int m = t - b * NUM_NODES;
    const float* base = logits + (size_t)b * MAX_VISIT * NUM_NODES + m;
    float vals[MAX_VISIT];
    float mx = -3.4e38f;
#pragma unroll
    for (int v = 0; v < MAX_VISIT; ++v) {
        vals[v] = base[(size_t)v * NUM_NODES];
        mx = fmaxf(mx, vals[v]);
    }
    float den = 0.f, num = 0.f;
#pragma unroll
    for (int v = 0; v < MAX_VISIT; ++v) {
        const float e = __expf(vals[v] - mx);
        den += e;
        num += e * beta[b * MAX_VISIT + v];
    }
    attn[t] = num / den;
}

// ---------------------------------------------------------------------------
// Per-edge BAT message + scatter-add (native global_atomic_add_f32).
// ---------------------------------------------------------------------------
__global__ void __launch_bounds__(256)
edge_kernel(const float* __restrict__ x, const float* __restrict__ efeat,
            const int* __restrict__ src, const int* __restrict__ dst,
            const int* __restrict__ node_ids, const int* __restrict__ batch,
            const float* __restrict__ attn, const float* __restrict__ wrw,
            const float* __restrict__ wrb, float* __restrict__ agg)
{
    const int lane = threadIdx.x & 31;
    const int gwave = (blockIdx.x * blockDim.x + threadIdx.x) >> 5;
    const int nwaves = (gridDim.x * blockDim.x) >> 5;
    const int j0 = lane * 4;
    const float4 w4 = *(const float4*)(wrw + j0);
    const float wb = wrb[0];
    for (int e = gwave; e < E_TOTAL; e += nwaves) {
        const int s = src[e];
        const int d = dst[e];
        const float4 ev = *(const float4*)(efeat + (size_t)e * HID + j0);
        const float4 xv = *(const float4*)(x + (size_t)s * HID + j0);
        float wr = w4.x * ev.x + w4.y * ev.y + w4.z * ev.z + w4.w * ev.w;
#pragma unroll
        for (int m = 16; m > 0; m >>= 1) wr += __shfl_xor(wr, m, 32);
        wr += wb;
        const float ae = attn[(size_t)batch[s] * NUM_NODES + node_ids[s]];
        float* ag = agg + (size_t)d * HID + j0;
        unsafeAtomicAdd(ag + 0, fmaxf(xv.x * ae + wr * ev.x, 0.f));
        unsafeAtomicAdd(ag + 1, fmaxf(xv.y * ae + wr * ev.y, 0.f));
        unsafeAtomicAdd(ag + 2, fmaxf(xv.z * ae + wr * ev.z, 0.f));
        unsafeAtomicAdd(ag + 3, fmaxf(xv.w * ae + wr * ev.w, 0.f));
    }
}

// ---------------------------------------------------------------------------
// Segment pooling accumulators.
// ---------------------------------------------------------------------------
__global__ void __launch_bounds__(256)
pool_kernel(const float* __restrict__ x, const int* __restrict__ batch,
            const int* __restrict__ node_ids, const float* __restrict__ ehr,
            float* __restrict__ gsum, float* __restrict__ nsum,
            float* __restrict__ cnt, float* __restrict__ mcnt)
{
    const int lane = threadIdx.x & 31;
    const int i = (blockIdx.x * blockDim.x + threadIdx.x) >> 5;
    if (i >= N_TOTAL) return;
    const int b = batch[i];
    const float mask = ehr[(size_t)b * NUM_NODES + node_ids[i]];
    const int j0 = lane * 4;
    const float4 xv = *(const float4*)(x + (size_t)i * HID + j0);
    float* g = gsum + b * HID + j0;
    unsafeAtomicAdd(g + 0, xv.x);
    unsafeAtomicAdd(g + 1, xv.y);
    unsafeAtomicAdd(g + 2, xv.z);
    unsafeAtomicAdd(g + 3, xv.w);
    if (mask != 0.f) {
        float* n = nsum + b * HID + j0;
        unsafeAtomicAdd(n + 0, xv.x);
        unsafeAtomicAdd(n + 1, xv.y);
        unsafeAtomicAdd(n + 2, xv.z);
        unsafeAtomicAdd(n + 3, xv.w);
    }
    if (lane == 0) {
        unsafeAtomicAdd(cnt + b, 1.f);
        unsafeAtomicAdd(mcnt + b, mask);
    }
}

// ---------------------------------------------------------------------------
// Final joint MLP.
// ---------------------------------------------------------------------------
__global__ void __launch_bounds__(256)
mlp_kernel(const float* __restrict__ gsum, const float* __restrict__ nsum,
           const float* __restrict__ cnt, const float* __restrict__ mcnt,
           const float* __restrict__ w1, const float* __restrict__ b1,
           const float* __restrict__ w2, const float* __restrict__ b2,
           float* __restrict__ out)
{
    __shared__ float hv[2 * HID];
    __shared__ float h1[HID];
    const int b = blockIdx.x;
    const int t = threadIdx.x;
    const float cg = fmaxf(cnt[b], 1.f);
    const float cm = fmaxf(mcnt[b], 1.f);
    hv[t] = (t < HID) ? gsum[b * HID + t] / cg : nsum[b * HID + (t - HID)] / cm;
    __syncthreads();
    if (t < HID) {
        float s = b1[t];
        const float* w = w1 + t * 2 * HID;
        for (int k = 0; k < 2 * HID; ++k) s += w[k] * hv[k];
        h1[t] = fmaxf(s, 0.f);
    }
    __syncthreads();
    if (t < OUT_DIM) {
        float s = b2[t];
        const float* w = w2 + t * HID;
        for (int k = 0; k < HID; ++k) s += w[k] * h1[k];
        out[b * OUT_DIM + t] = s;
    }
}

extern "C" void kernel_launch(void* const* d_in, const int* in_sizes, int n_in,
                              void* d_out, int out_size, void* d_ws, size_t ws_size,
                              hipStream_t stream) {
    const float* node_table = (const float*)d_in[0];   // [5000,128]
    const float* rel_table  = (const float*)d_in[1];   // [100,128]
    const float* lin_w      = (const float*)d_in[2];   // [128,128]
    const float* lin_b      = (const float*)d_in[3];   // [128]
    const float* alpha_w    = (const float*)d_in[4];   // [3,5000,5000]
    const float* alpha_b    = (const float*)d_in[5];   // [3,5000]
    const float* beta_w     = (const float*)d_in[6];   // [3,1,5000]
    const float* beta_b     = (const float*)d_in[7];   // [3,1]
    const float* wr_w       = (const float*)d_in[8];   // [3,1,128]
    const float* wr_b       = (const float*)d_in[9];   // [3,1]
    const float* conv_w     = (const float*)d_in[10];  // [3,128,128]
    const float* conv_b     = (const float*)d_in[11];  // [3,128]
    const float* mlp_w1     = (const float*)d_in[12];  // [128,256]
    const float* mlp_b1     = (const float*)d_in[13];  // [128]
    const float* mlp_w2     = (const float*)d_in[14];  // [25,128]
    const float* mlp_b2     = (const float*)d_in[15];  // [25]
    const float* visit      = (const float*)d_in[16];  // [32,20,5000]
    const float* ehr        = (const float*)d_in[17];  // [32,5000]
    const int*   node_ids   = (const int*)d_in[18];    // [100000]
    const int*   rel_ids    = (const int*)d_in[19];    // [400000]
    const int*   edge_index = (const int*)d_in[20];    // [2,400000]
    const int*   batch      = (const int*)d_in[21];    // [100000]

    const int* src = edge_index;
    const int* dst = edge_index + E_TOTAL;

    float* ws = (float*)d_ws;
    size_t off = 0;
    float* x0    = ws + off; off += (size_t)N_TOTAL * HID;
    float* x1    = ws + off; off += (size_t)N_TOTAL * HID;
    float* ef    = ws + off; off += (size_t)E_TOTAL * HID;
    float* alog  = ws + off; off += (size_t)BATCH * MAX_VISIT * NUM_NODES;
    float* betav = ws + off; off += (size_t)BATCH * MAX_VISIT;
    float* attn  = ws + off; off += (size_t)BATCH * NUM_NODES;
    float* agg   = ws + off; off += (size_t)N_TOTAL * HID;
    float* gsum  = ws + off; off += (size_t)BATCH * HID;
    float* nsum  = ws + off; off += (size_t)BATCH * HID;
    float* cnt   = ws + off; off += BATCH;
    float* mcnt  = ws + off; off += BATCH;

    gemm128_gather_wmma<<<dim3(N_TOTAL / 16, 2), 128, 0, stream>>>(
        node_table, node_ids, lin_w, lin_b, x0);
    gemm128_gather_wmma<<<dim3(E_TOTAL / 16, 2), 128, 0, stream>>>(
        rel_table, rel_ids, lin_w, lin_b, ef);

    for (int l = 0; l < LAYERS; ++l) {
        float* xin  = (l & 1) ? x1 : x0;
        float* xout = (l & 1) ? x0 : x1;

        alpha_gemm_wmma<<<dim3(BATCH * MAX_VISIT / 64, (NUM_NODES + 15) / 16),
                          128, 0, stream>>>(
            visit, alpha_w + (size_t)l * NUM_NODES * NUM_NODES,
            alpha_b + (size_t)l * NUM_NODES, alog);

        beta_kernel<<<BATCH * MAX_VISIT, 32, 0, stream>>>(
            visit, beta_w + (size_t)l * NUM_NODES, beta_b + l, betav);

        attn_kernel<<<(BATCH * NUM_NODES + 255) / 256, 256, 0, stream>>>(
            alog, betav, attn);

        (void)hipMemsetAsync(agg, 0, (size_t)N_TOTAL * HID * sizeof(float), stream);

        edge_kernel<<<4096, 256, 0, stream>>>(
            xin, ef, src, dst, node_ids, batch, attn,
            wr_w + (size_t)l * HID, wr_b + l, agg);

        gemm128_add_wmma<<<dim3(N_TOTAL / 16, 2), 128, 0, stream>>>(
            agg, xin, conv_w + (size_t)l * HID * HID,
            conv_b + (size_t)l * HID, xout);
    }

    float* xfin = (LAYERS & 1) ? x1 : x0;

    (void)hipMemsetAsync(gsum, 0, (2 * BATCH * HID + 2 * BATCH) * sizeof(float), stream);

    pool_kernel<<<N_TOTAL / 8, 256, 0, stream>>>(
        xfin, batch, node_ids, ehr, gsum, nsum, cnt, mcnt);

    mlp_kernel<<<BATCH, 256, 0, stream>>>(
        gsum, nsum, cnt, mcnt, mlp_w1, mlp_b1, mlp_w2, mlp_b2, (float*)d_out);
}